// LinearCRF_43508018709169
// MI455X (gfx1250) — compile-verified
//
#include <hip/hip_runtime.h>
#include <hip/hip_bf16.h>
#include <stdint.h>

// Problem sizes (fixed by the reference module)
#define B_   4096
#define T_   2048
// L == 2 (hardcoded)

#define ROWS 64     // batch rows per workgroup (= blockDim.x, 2 wave32s)
#define TT   32     // timesteps per LDS tile
#define NT   (T_ / TT)
#define FSTR 68     // floats per LDS feats row: TT*2 data + 4 pad (272B: 16B-aligned, 2-way max conflict)
#define MSTR 48     // bytes per LDS mask row: TT data + 16 pad (16B-aligned)

typedef unsigned int v4u __attribute__((ext_vector_type(4)));
typedef int          v8i __attribute__((ext_vector_type(8)));
typedef int          v4i __attribute__((ext_vector_type(4)));

// log-sum-exp of two values: max + log(1 + exp(min - max))
__device__ __forceinline__ float lse2(float a, float b) {
    float mx = fmaxf(a, b);
    float mn = fminf(a, b);
    return mx + __logf(1.0f + __expf(mn - mx));
}

// ---------------------------------------------------------------------------
// TDM: one tensor_load_to_lds moves a [64 rows x 256B] 2-D strided tile of
// feats (row stride 16KB in HBM) into LDS, inserting 16B of pad after every
// 256B row in hardware (pad_interval=64 DWORDs, pad_amount=4 DWORDs) so the
// LDS row stride is FSTR*4 = 272B. Issued by one wave; tracked by TENSORcnt.
// ---------------------------------------------------------------------------
__device__ __forceinline__ void tdm_load_tile(const float* feats, int b0, int t0,
                                              unsigned lds_off) {
    unsigned long long ga = (unsigned long long)(const void*)
        (feats + (size_t)b0 * (size_t)(T_ * 2) + (size_t)t0 * 2);
    v4u g0;
    g0.x = 1u;                                      // count=1 (valid user descriptor)
    g0.y = lds_off;                                 // lds_addr
    g0.z = (unsigned)(ga & 0xFFFFFFFFull);          // global_addr[31:0]
    g0.w = (unsigned)((ga >> 32) & 0x01FFFFFFull)   // global_addr[56:32]
         | 0x80000000u;                             // type = 2 ("image")
    v8i g1;
    g1[0] = (int)((2u << 16)      // data_size = 4 bytes
                | (1u << 20)      // pad_enable
                | (5u << 22)      // pad_interval: every 64 DWORDs (256B)
                | (3u << 25));    // pad_amount: 4 DWORDs (16B)
    g1[1] = (int)(((unsigned)(T_ * 2) & 0xFFFFu) << 16);   // tensor_dim0[15:0]
    g1[2] = (int)((((unsigned)(T_ * 2) >> 16) & 0xFFFFu)   // tensor_dim0[31:16]
                | (((unsigned)B_ & 0xFFFFu) << 16));       // tensor_dim1[15:0]
    g1[3] = (int)(((unsigned)B_ >> 16) & 0xFFFFu)          // tensor_dim1[31:16]
          | (int)((unsigned)(TT * 2) << 16);               // tile_dim0 = 64 elems
    g1[4] = ROWS;                                          // tile_dim1 = 64 rows (tile_dim2 = 0)
    g1[5] = T_ * 2;                                        // tensor_dim0_stride = 4096 elems
    g1[6] = 0;
    g1[7] = 0;
    v4i gz = {0, 0, 0, 0};
#if defined(__clang_major__) && (__clang_major__ >= 23)
    v8i gz8 = {0, 0, 0, 0, 0, 0, 0, 0};
    __builtin_amdgcn_tensor_load_to_lds(g0, g1, gz, gz, gz8, 0);
#else
    __builtin_amdgcn_tensor_load_to_lds(g0, g1, gz, gz, 0);
#endif
}

// Stage one tile: feats via TDM (wave 0 only), mask rows via async global->LDS
// (2 x b128 granules per thread, coalesced; tracked by ASYNCcnt).
__device__ __forceinline__ void stage_tile(const float* feats, const unsigned char* mask,
                                           int b0, int t0, unsigned flds, unsigned mlds,
                                           int tid) {
    if (tid < 32) tdm_load_tile(feats, b0, t0, flds);
    const char* mb = (const char*)mask;
#pragma unroll
    for (int k = 0; k < 2; ++k) {
        int g   = k * 64 + tid;
        int row = g >> 1;           // 2 granules (32B) per mask row
        int col = (g & 1) << 4;
        unsigned loff = mlds + (unsigned)(row * MSTR + col);
        unsigned long long gaddr = (unsigned long long)(mb
            + (size_t)(b0 + row) * (size_t)T_ + (size_t)t0 + (size_t)col);
        asm volatile("global_load_async_to_lds_b128 %0, %1, off"
                     :: "v"(loff), "v"(gaddr) : "memory");
    }
}

__device__ __forceinline__ void wait_staged(bool prefetched) {
    if (prefetched) {
        __builtin_amdgcn_s_wait_tensorcnt(1);                 // oldest TDM tile done
        asm volatile("s_wait_asynccnt 0x2" ::: "memory");     // oldest 2 mask granule ops done
    } else {
        __builtin_amdgcn_s_wait_tensorcnt(0);
        asm volatile("s_wait_asynccnt 0x0" ::: "memory");
    }
}

// One thread per batch row. Blocks [0, B/ROWS) run the forward scan (and Z),
// blocks [B/ROWS, 2*B/ROWS) run the backward scan concurrently (halves the
// serial critical path; only the final marginal combine needs both + Z).
__global__ __launch_bounds__(ROWS) void crf_scan_kernel(
    const float* __restrict__ feats, const unsigned char* __restrict__ mask,
    const float* __restrict__ trans,
    float* __restrict__ fwd, float* __restrict__ bwd, float* __restrict__ Zb)
{
    __shared__ __align__(16) float         ftile[2][ROWS * FSTR];
    __shared__ __align__(16) unsigned char mtile[2][ROWS * MSTR];

    const int tid = threadIdx.x;
    const bool is_bwd = blockIdx.x >= (B_ / ROWS);
    const int b0 = (int)(is_bwd ? blockIdx.x - (B_ / ROWS) : blockIdx.x) * ROWS;
    const int b  = b0 + tid;

    const unsigned flds[2] = { (unsigned)(unsigned long long)(const void*)&ftile[0][0],
                               (unsigned)(unsigned long long)(const void*)&ftile[1][0] };
    const unsigned mlds[2] = { (unsigned)(unsigned long long)(const void*)&mtile[0][0],
                               (unsigned)(unsigned long long)(const void*)&mtile[1][0] };

    // transitions (L,L) row-major: t[i][j]
    const float t00 = trans[0], t01 = trans[1], t10 = trans[2], t11 = trans[3];

    float p0 = 0.f, p1 = 0.f;
    bool mprev = false;

    // Prologue: stage first tile (tile index in scan order 0).
    {
        const int tile0 = is_bwd ? (NT - 1) : 0;
        stage_tile(feats, mask, b0, tile0 * TT, flds[0], mlds[0], tid);
    }

    for (int i = 0; i < NT; ++i) {
        const int cur = is_bwd ? (NT - 1 - i) : i;
        const bool havenext = (i + 1 < NT);
        if (havenext) {
            const int nxt = is_bwd ? (cur - 1) : (cur + 1);
            stage_tile(feats, mask, b0, nxt * TT, flds[(i + 1) & 1], mlds[(i + 1) & 1], tid);
        }
        wait_staged(havenext);
        __syncthreads();   // all waves' staged data visible

        const int bi = i & 1;
        const float* frow = &ftile[bi][tid * FSTR];
        const unsigned char* mrow = &mtile[bi][tid * MSTR];
        const int t0 = cur * TT;

        if (!is_bwd) {
            // ---------- forward: alpha_t ----------
#pragma unroll 4
            for (int tt = 0; tt < TT; ++tt) {
                const int t = t0 + tt;
                float f0 = frow[tt * 2], f1 = frow[tt * 2 + 1];
                if (t == 0) {
                    p0 = f0; p1 = f1;
                } else {
                    // cp[j] = f[j] + LSE_i(p[i] + trans[i][j])
                    float cp0 = f0 + lse2(p0 + t00, p1 + t10);
                    float cp1 = f1 + lse2(p0 + t01, p1 + t11);
                    if (mrow[tt]) { p0 = cp0; p1 = cp1; }
                }
                fwd[((size_t)b * T_ + t) * 2    ] = p0;
                fwd[((size_t)b * T_ + t) * 2 + 1] = p1;
            }
        } else {
            // ---------- backward: beta_t ----------
#pragma unroll 4
            for (int tt = TT - 1; tt >= 0; --tt) {
                const int t = t0 + tt;
                float f0 = frow[tt * 2], f1 = frow[tt * 2 + 1];
                bool m = mrow[tt] != 0;
                if (t == T_ - 1) {
                    p0 = m ? f0 : 0.f;
                    p1 = m ? f1 : 0.f;
                } else {
                    // cp[j] = f[j] + LSE_i(p[i] + trans[j][i])
                    float cp0 = f0 + lse2(p0 + t00, p1 + t01);
                    float cp1 = f1 + lse2(p0 + t10, p1 + t11);
                    cp0 = mprev ? cp0 : f0;
                    cp1 = mprev ? cp1 : f1;
                    if (m) { p0 = cp0; p1 = cp1; }
                }
                mprev = m;
                bwd[((size_t)b * T_ + t) * 2    ] = p0;
                bwd[((size_t)b * T_ + t) * 2 + 1] = p1;
            }
        }
        __syncthreads();   // tile consumed before the buffer is re-staged
    }

    if (!is_bwd) Zb[b] = lse2(p0, p1);
}

// Fully parallel, coalesced float4 marginal combine: out = exp(fwd+bwd-feats-Z[b])
__global__ __launch_bounds__(256) void crf_marginal_kernel(
    const float4* __restrict__ feats, const float4* __restrict__ fwd,
    const float4* __restrict__ bwd, const float* __restrict__ Zb,
    float4* __restrict__ out)
{
    const size_t i = (size_t)blockIdx.x * 256 + threadIdx.x;  // float4 index
    const int b = (int)(i >> 10);  // T*2/4 = 1024 float4 per batch row
    const float z = Zb[b];
    float4 f = feats[i], a = fwd[i], c = bwd[i];
    float4 o;
    o.x = __expf(a.x + c.x - f.x - z);
    o.y = __expf(a.y + c.y - f.y - z);
    o.z = __expf(a.z + c.z - f.z - z);
    o.w = __expf(a.w + c.w - f.w - z);
    out[i] = o;
}

extern "C" void kernel_launch(void* const* d_in, const int* in_sizes, int n_in,
                              void* d_out, int out_size, void* d_ws, size_t ws_size,
                              hipStream_t stream) {
    (void)in_sizes; (void)n_in; (void)out_size; (void)ws_size;
    const float*         feats = (const float*)d_in[0];          // (B,T,2) f32
    const unsigned char* mask  = (const unsigned char*)d_in[1];  // (B,T) bool (1 byte)
    const float*         trans = (const float*)d_in[2];          // (2,2) f32

    // Workspace layout (needs 2*B*T*2*4 + B*4 ~= 128 MB; L2-resident at 192 MB)
    float* fwd = (float*)d_ws;
    float* bwd = fwd + (size_t)B_ * T_ * 2;
    float* Zb  = bwd + (size_t)B_ * T_ * 2;

    // Forward and backward scans run concurrently (independent halves of grid).
    crf_scan_kernel<<<dim3(2 * (B_ / ROWS)), dim3(ROWS), 0, stream>>>(
        feats, mask, trans, fwd, bwd, Zb);

    // Elementwise marginals (coalesced float4).
    const int n4 = (B_ * T_ * 2) / 4;
    crf_marginal_kernel<<<dim3(n4 / 256), dim3(256), 0, stream>>>(
        (const float4*)feats, (const float4*)fwd, (const float4*)bwd, Zb,
        (float4*)d_out);
}